// ChamferLoss_52725018526341
// MI455X (gfx1250) — compile-verified
//
#include <hip/hip_runtime.h>

typedef __attribute__((ext_vector_type(2))) float v2f;
typedef __attribute__((ext_vector_type(4))) float v4f;
typedef __attribute__((ext_vector_type(8))) float v8f;

#define B_ 16
#define N_ 4096

// ---------------------------------------------------------------------------
// Kernel 1: normalize each 3-vector point and write it padded to 4 floats
// (4th component = 0 so K=4 WMMA sees an exact 3-dot-product).
// Input layout: [B, N, 3] f32 contiguous. Output: [B, N, 4] f32.
// ---------------------------------------------------------------------------
__global__ void __launch_bounds__(256)
chamfer_normalize(const float* __restrict__ a,
                  const float* __restrict__ b,
                  float* __restrict__ ahat,
                  float* __restrict__ bhat) {
    const int i = blockIdx.x * blockDim.x + threadIdx.x;   // 0 .. B*N-1
    const float* __restrict__ src = (blockIdx.y == 0) ? a : b;
    float* __restrict__ dst       = (blockIdx.y == 0) ? ahat : bhat;

    const size_t s3 = (size_t)i * 3;
    const float x = src[s3 + 0];
    const float y = src[s3 + 1];
    const float z = src[s3 + 2];
    const float r = rsqrtf(x * x + y * y + z * z);   // ref divides by norm; rsqrt equiv.
    v4f o;
    o.x = x * r; o.y = y * r; o.z = z * r; o.w = 0.0f;
    *(v4f*)(dst + ((size_t)i << 2)) = o;             // global_store_b128
}

// ---------------------------------------------------------------------------
// Kernel 2: one wave owns one (direction, batch, 16-row tile).
// Inner loop over 256 column tiles: v_wmma_f32_16x16x4_f32 computes the
// 16x16 cosine-similarity tile directly (C = 0), then elementwise running
// max. A-operand is loop-invariant (held in 2 VGPRs).
//
// WMMA f32 16x16x4 operand layout (wave32):
//   A (16x4): lanes 0-15 -> M=lane, VGPR0=K0 VGPR1=K1; lanes 16-31 -> K2,K3.
//   B (4x16): lanes 0-15 -> N=lane, VGPR0=K0 VGPR1=K1; lanes 16-31 -> K2,K3.
// With the N x 4 padded row-major point array, BOTH operands are a single
// b64 load of 2 consecutive floats at point[base + lane%16][2*(lane/16)].
//   D (16x16): VGPR v, lane L -> row = v + 8*(L/16), col = L%16.
// ---------------------------------------------------------------------------
__global__ void __launch_bounds__(256)
chamfer_rowmax(const float* __restrict__ ahat,
               const float* __restrict__ bhat,
               float* __restrict__ out,
               float scale) {
    const int lane = threadIdx.x & 31;
    const int wave = threadIdx.x >> 5;
    const int half = lane >> 4;        // 0: K0/K1 slice, 1: K2/K3 slice
    const int lrow = lane & 15;

    const unsigned bid = blockIdx.x;           // 1024 blocks = 2 dirs * 16 b * 32 groups
    const int dir     = bid >> 9;
    const int batch   = (bid >> 5) & 15;
    const int tilegrp = bid & 31;
    const int rowbase = (tilegrp * 8 + wave) * 16;

    const float* __restrict__ rows = dir ? bhat : ahat;
    const float* __restrict__ cols = dir ? ahat : bhat;

    // Loop-invariant A operand: 2 floats of unit-row (rowbase+lrow), K-slice `half`.
    const float* rowPtr =
        rows + ((((size_t)batch * N_ + rowbase + lrow) << 2) + (half << 1));
    const v2f aop = *(const v2f*)rowPtr;

    // Column base for this lane within the batch; tile j is +64 floats apart.
    const float* colBase =
        cols + ((((size_t)batch * N_ + lrow) << 2) + (half << 1));

    v8f acc;
#pragma unroll
    for (int v = 0; v < 8; ++v) acc[v] = -3.0e38f;

#pragma unroll 4
    for (int j = 0; j < N_ / 16; ++j) {
        const v2f bop = *(const v2f*)(colBase + (size_t)j * 64);  // 16 pts * 4 f32
        v8f c = {0.f, 0.f, 0.f, 0.f, 0.f, 0.f, 0.f, 0.f};
        // D = A x B (+0): 16x16 tile of cosine similarities.
        v8f d = __builtin_amdgcn_wmma_f32_16x16x4_f32(
            /*neg_a=*/false, aop, /*neg_b=*/false, bop,
            /*c_mod=*/(short)0, c, /*reuse_a=*/false, /*reuse_b=*/false);
#pragma unroll
        for (int v = 0; v < 8; ++v) acc[v] = fmaxf(acc[v], d[v]);
    }

    // Max across the 16 lanes of each half (cols 0..15 live on lanes L%16).
#pragma unroll
    for (int v = 0; v < 8; ++v) {
        float x = acc[v];
        x = fmaxf(x, __shfl_xor(x, 8, 32));
        x = fmaxf(x, __shfl_xor(x, 4, 32));
        x = fmaxf(x, __shfl_xor(x, 2, 32));
        x = fmaxf(x, __shfl_xor(x, 1, 32));
        acc[v] = x;   // rowmax for row (v + 8*half) of this tile, all lanes of half
    }

    // Epilogue: sum (1 - min(1, rowmax))^2 over the 16 rows of the tile.
    float s = 0.0f;
#pragma unroll
    for (int v = 0; v < 8; ++v) {
        const float r = fminf(acc[v], 1.0f);
        const float t = 1.0f - r;
        s += t * t;   // lanes 0-15: rows v; lanes 16-31: rows v+8
    }
    s += __shfl_xor(s, 16, 32);   // combine the two halves
    if (lane == 0) atomicAdd(out, s * scale);
}

// ---------------------------------------------------------------------------
extern "C" void kernel_launch(void* const* d_in, const int* in_sizes, int n_in,
                              void* d_out, int out_size, void* d_ws, size_t ws_size,
                              hipStream_t stream) {
    const float* m1 = (const float*)d_in[0];   // [16, 4096, 3] f32
    const float* m2 = (const float*)d_in[1];   // [16, 4096, 3] f32
    float* out = (float*)d_out;                // scalar loss

    float* ahat = (float*)d_ws;                       // [B, N, 4] f32 (1 MB)
    float* bhat = ahat + (size_t)B_ * N_ * 4;         // [B, N, 4] f32 (1 MB)

    hipMemsetAsync(d_out, 0, sizeof(float), stream);  // graph-capturable

    dim3 gN((B_ * N_) / 256, 2);
    chamfer_normalize<<<gN, 256, 0, stream>>>(m1, m2, ahat, bhat);

    // 2 dirs * 16 batches * 32 tile-groups = 1024 blocks, 8 waves each
    // (one wave per 16-row tile). scale folds the /n and /b of the reference.
    const float scale = 1.0f / ((float)N_ * (float)B_);
    chamfer_rowmax<<<1024, 256, 0, stream>>>(ahat, bhat, out, scale);
}